// GCNConvLayer_13048110645761
// MI455X (gfx1250) — compile-verified
//
#include <hip/hip_runtime.h>
#include <hip/hip_bf16.h>

typedef __attribute__((ext_vector_type(2))) float v2f;
typedef __attribute__((ext_vector_type(8))) float v8f;

#define DIM 64

// ---------------------------------------------------------------------------
// Kernel 1: init degrees to 1.0 (self loop)
// ---------------------------------------------------------------------------
__global__ __launch_bounds__(256) void k_init_deg(float* __restrict__ deg, int N) {
    int i = blockIdx.x * blockDim.x + threadIdx.x;
    if (i < N) deg[i] = 1.0f;
}

// ---------------------------------------------------------------------------
// Kernel 2: zero the aggregation accumulator (lives in d_out)
// ---------------------------------------------------------------------------
__global__ __launch_bounds__(256) void k_zero4(float4* __restrict__ p, int n4) {
    int i = blockIdx.x * blockDim.x + threadIdx.x;
    if (i < n4) p[i] = make_float4(0.f, 0.f, 0.f, 0.f);
}

// ---------------------------------------------------------------------------
// Kernel 3: degree accumulation: deg[dst] += 1 over all edges
// edge_index is int64 [2, E]: row0 = src, row1 = dst
// ---------------------------------------------------------------------------
__global__ __launch_bounds__(256) void k_degree(const long long* __restrict__ ei,
                                                float* __restrict__ deg, int E) {
    int e = blockIdx.x * blockDim.x + threadIdx.x;
    if (e < E) {
        int dst = (int)ei[(long long)E + e];
        unsafeAtomicAdd(&deg[dst], 1.0f);
    }
}

// ---------------------------------------------------------------------------
// Kernel 4: dinv = rsqrt(deg) (in place)
// ---------------------------------------------------------------------------
__global__ __launch_bounds__(256) void k_rsqrt(float* __restrict__ deg, int N) {
    int i = blockIdx.x * blockDim.x + threadIdx.x;
    if (i < N) deg[i] = rsqrtf(deg[i]);
}

// ---------------------------------------------------------------------------
// Kernel 5: h = x @ W via V_WMMA_F32_16X16X4_F32 (fp32, no precision loss)
// One wave computes a 16-row stripe of h (16 x 64), i.e. four 16x16 tiles,
// accumulating over K=64 in steps of 4.
//
// A fragment (16x4 f32): lanes 0-15: M=lane, K={k,k+1}; lanes 16-31: M=lane-16,
// K={k+2,k+3}.  B fragment (4x16 f32): lanes 0-15: N=lane, K={k,k+1};
// lanes 16-31: N=lane-16, K={k+2,k+3}.
// C/D (16x16 f32): VGPR r -> M = r + 8*(lane>=16), N = lane%16.
// ---------------------------------------------------------------------------
__global__ __launch_bounds__(256) void k_gemm_wmma(const float* __restrict__ x,
                                                   const float* __restrict__ W,
                                                   float* __restrict__ h, int N) {
    const int wave = (blockIdx.x * blockDim.x + threadIdx.x) >> 5;
    const int lane = threadIdx.x & 31;
    const int row0 = wave * 16;
    if (row0 >= N) return;

    const int half = lane >> 4;      // 0: lanes 0-15, 1: lanes 16-31
    const int lmod = lane & 15;

    // clamp for (impossible here: N%16==0) partial tiles; stores are guarded
    const int arow = min(row0 + lmod, N - 1);
    const float* __restrict__ xrow = x + (size_t)arow * DIM;

    v8f acc0 = {}, acc1 = {}, acc2 = {}, acc3 = {};

    #pragma unroll
    for (int k = 0; k < DIM; k += 4) {
        const int ka = k + half * 2;          // this lane's K base
        v2f a;
        a.x = xrow[ka];
        a.y = xrow[ka + 1];

        v2f b0, b1, b2, b3;
        const float* __restrict__ Wr0 = W + (size_t)ka * DIM + lmod;
        const float* __restrict__ Wr1 = Wr0 + DIM;
        b0.x = Wr0[0];  b0.y = Wr1[0];
        b1.x = Wr0[16]; b1.y = Wr1[16];
        b2.x = Wr0[32]; b2.y = Wr1[32];
        b3.x = Wr0[48]; b3.y = Wr1[48];

        acc0 = __builtin_amdgcn_wmma_f32_16x16x4_f32(false, a, false, b0,
                                                     (short)0, acc0, false, false);
        acc1 = __builtin_amdgcn_wmma_f32_16x16x4_f32(false, a, false, b1,
                                                     (short)0, acc1, false, false);
        acc2 = __builtin_amdgcn_wmma_f32_16x16x4_f32(false, a, false, b2,
                                                     (short)0, acc2, false, false);
        acc3 = __builtin_amdgcn_wmma_f32_16x16x4_f32(false, a, false, b3,
                                                     (short)0, acc3, false, false);
    }

    // store D: element r of acc -> row (row0 + r + 8*half), col (tile*16 + lmod)
    #pragma unroll
    for (int r = 0; r < 8; ++r) {
        const int row = row0 + r + 8 * half;
        if (row < N) {
            float* __restrict__ hp = h + (size_t)row * DIM + lmod;
            hp[0]  = acc0[r];
            hp[16] = acc1[r];
            hp[32] = acc2[r];
            hp[48] = acc3[r];
        }
    }
}

// ---------------------------------------------------------------------------
// Kernel 6: edge aggregation: agg[dst] += h[src] * dinv[src]*dinv[dst]
// 16 lanes cooperate on one edge; each lane covers 4 consecutive channels
// (float4 gather + 4 non-returning fp32 atomics). h and agg (25.6 MB each)
// both live in the 192 MB L2, so this random traffic stays on-chip.
// ---------------------------------------------------------------------------
__global__ __launch_bounds__(256) void k_edge_agg(const long long* __restrict__ ei,
                                                  const float* __restrict__ h,
                                                  const float* __restrict__ dinv,
                                                  float* __restrict__ agg, int E) {
    const int tid = blockIdx.x * blockDim.x + threadIdx.x;
    const int e = tid >> 4;
    if (e >= E) return;
    const int c = (tid & 15) * 4;

    const int src = (int)ei[e];
    const int dst = (int)ei[(long long)E + e];
    const float norm = dinv[src] * dinv[dst];

    const float4 hv = *(const float4*)(h + (size_t)src * DIM + c);
    float* __restrict__ ap = agg + (size_t)dst * DIM + c;
    unsafeAtomicAdd(ap + 0, hv.x * norm);
    unsafeAtomicAdd(ap + 1, hv.y * norm);
    unsafeAtomicAdd(ap + 2, hv.z * norm);
    unsafeAtomicAdd(ap + 3, hv.w * norm);
}

// ---------------------------------------------------------------------------
// Kernel 7: epilogue: out = x + relu(agg + h*dinv^2 + b)   (agg is in d_out)
// ---------------------------------------------------------------------------
__global__ __launch_bounds__(256) void k_epilogue(const float* __restrict__ x,
                                                  const float* __restrict__ h,
                                                  const float* __restrict__ dinv,
                                                  const float* __restrict__ bias,
                                                  float* __restrict__ out, int N) {
    const int tid = blockIdx.x * blockDim.x + threadIdx.x;
    const int node = tid >> 4;
    if (node >= N) return;
    const int c = (tid & 15) * 4;

    const float di = dinv[node];
    const float selfn = di * di;

    const size_t idx = (size_t)node * DIM + c;
    const float4 a  = *(const float4*)(out + idx);   // agg
    const float4 hv = *(const float4*)(h + idx);
    const float4 xv = *(const float4*)(x + idx);
    const float4 bv = *(const float4*)(bias + c);

    float4 r;
    r.x = fmaxf(a.x + hv.x * selfn + bv.x, 0.f) + xv.x;
    r.y = fmaxf(a.y + hv.y * selfn + bv.y, 0.f) + xv.y;
    r.z = fmaxf(a.z + hv.z * selfn + bv.z, 0.f) + xv.z;
    r.w = fmaxf(a.w + hv.w * selfn + bv.w, 0.f) + xv.w;
    *(float4*)(out + idx) = r;
}

// ---------------------------------------------------------------------------
extern "C" void kernel_launch(void* const* d_in, const int* in_sizes, int n_in,
                              void* d_out, int out_size, void* d_ws, size_t ws_size,
                              hipStream_t stream) {
    const float*     x  = (const float*)d_in[0];
    const long long* ei = (const long long*)d_in[1];
    const float*     W  = (const float*)d_in[2];
    const float*     b  = (const float*)d_in[3];
    float* out = (float*)d_out;

    const int N = in_sizes[0] / DIM;        // 100000
    const int E = in_sizes[1] / 2;          // 1600000

    // workspace layout (floats): [dinv: N (padded)] [h: N*DIM]
    float* wsf  = (float*)d_ws;
    float* dinv = wsf;
    float* h    = wsf + (((size_t)N + 63) / 64) * 64;

    const int B256 = 256;

    // 1. deg = 1.0
    k_init_deg<<<(N + B256 - 1) / B256, B256, 0, stream>>>(dinv, N);
    // 2. agg (d_out) = 0
    const int n4 = N * (DIM / 4);
    k_zero4<<<(n4 + B256 - 1) / B256, B256, 0, stream>>>((float4*)out, n4);
    // 3. deg[dst] += 1 over edges
    k_degree<<<(E + B256 - 1) / B256, B256, 0, stream>>>(ei, dinv, E);
    // 4. dinv = rsqrt(deg)
    k_rsqrt<<<(N + B256 - 1) / B256, B256, 0, stream>>>(dinv, N);
    // 5. h = x @ W  (WMMA fp32)
    const int waves  = (N + 15) / 16;
    const int gblocks = (waves + 7) / 8;    // 8 waves per 256-thread block
    k_gemm_wmma<<<gblocks, B256, 0, stream>>>(x, W, h, N);
    // 6. edge aggregation into d_out
    const long long ethreads = (long long)E * 16;
    k_edge_agg<<<(int)((ethreads + B256 - 1) / B256), B256, 0, stream>>>(ei, h, dinv, out, E);
    // 7. epilogue
    const long long nthreads = (long long)N * 16;
    k_epilogue<<<(int)((nthreads + B256 - 1) / B256), B256, 0, stream>>>(x, h, dinv, b, out, N);
}